// GCN_34282428957112
// MI455X (gfx1250) — compile-verified
//
#include <hip/hip_runtime.h>
#include <hip/hip_bf16.h>

typedef __attribute__((ext_vector_type(2))) float v2f;
typedef __attribute__((ext_vector_type(8))) float v8f;
typedef __attribute__((ext_vector_type(2))) unsigned long long v2u64;  // native vec -> NT-load ok

#define GCN_N_NODES 2000000
#define GCN_N_EDGES 32000000LL
#define GCN_N_OUT   400000   // N_NODES / 5

// ---------------------------------------------------------------------------
// Kernel 0: zero the 400K-float accumulator in workspace (harness poisons ws).
// ---------------------------------------------------------------------------
__global__ __launch_bounds__(256) void gcn_zero_agg(float* __restrict__ agg) {
    int i = blockIdx.x * blockDim.x + threadIdx.x;
    if (i < GCN_N_OUT) agg[i] = 0.0f;
}

// ---------------------------------------------------------------------------
// Kernel 1: edge scatter. Stream 512MB of int64 edge indices with b128
// non-temporal loads (+ prefetch ahead), filter dst%5==0, gather x[src]
// (x is 8MB -> L2 resident), scale by w_conv, global_atomic_add_f32 into the
// 1.6MB L2-resident accumulator indexed by dst/5.
// ---------------------------------------------------------------------------
__global__ __launch_bounds__(256) void gcn_edge_scatter(
    const long long* __restrict__ src_idx, const long long* __restrict__ dst_idx,
    const float* __restrict__ x, const float* __restrict__ w_conv,
    float* __restrict__ agg) {
    const float w = w_conv[0];
    const long long npair  = GCN_N_EDGES / 2;
    const long long stride = (long long)gridDim.x * blockDim.x;
    const v2u64* s2 = (const v2u64*)src_idx;
    const v2u64* d2 = (const v2u64*)dst_idx;

    for (long long p = (long long)blockIdx.x * blockDim.x + threadIdx.x;
         p < npair; p += stride) {
        long long pf = p + stride * 8;          // prefetch ~8 iterations ahead
        if (pf < npair) {
            __builtin_prefetch(&d2[pf], 0, 0);  // -> global_prefetch_b8
            __builtin_prefetch(&s2[pf], 0, 0);
        }
        v2u64 d = __builtin_nontemporal_load(&d2[p]);   // 16B NT stream
        v2u64 s = __builtin_nontemporal_load(&s2[p]);
        unsigned dA = (unsigned)d.x;
        unsigned dB = (unsigned)d.y;
        if (dA % 5u == 0u) atomicAdd(&agg[dA / 5u], x[(unsigned)s.x] * w);
        if (dB % 5u == 0u) atomicAdd(&agg[dB / 5u], x[(unsigned)s.y] * w);
    }
}

// ---------------------------------------------------------------------------
// Kernel 2: per-node MLP via V_WMMA_F32_16X16X4_F32.
//   a      = agg[node] + b_conv
//   h_k    = relu(a * (w1[0,k]+w1[1,k]) + b1[k])        (k = 0..3)
//   out    = sum_k h_k * w2[k] + b2
// Each wave handles 16 nodes: A = h (16x4 fp32), B column 0 = w2 (4x16),
// D[:,0] = 16 outputs. A-layout (ISA 7.12.2, 32-bit A 16x4): lanes 0-15 hold
// K={0,1} in VGPR{0,1}; lanes 16-31 hold K={2,3}. D column N=0 lives in
// lane 0 (M=0..7, VGPR0..7) and lane 16 (M=8..15).
// ---------------------------------------------------------------------------
__global__ __launch_bounds__(256) void gcn_mlp_wmma(
    const float* __restrict__ agg, const float* __restrict__ b_conv,
    const float* __restrict__ w1, const float* __restrict__ b1,
    const float* __restrict__ w2, const float* __restrict__ b2,
    float* __restrict__ out) {
    const int lane   = threadIdx.x & 31;
    const int waveId = (blockIdx.x * blockDim.x + threadIdx.x) >> 5;  // 0..24999
    const int base   = waveId * 16;
    const int m      = lane & 15;   // node row within the 16-group
    const int half   = lane >> 4;   // 0: K={0,1}, 1: K={2,3}

    const float bc = b_conv[0];
    // effective first-layer weights: concat([agg,agg]) @ w1  ==  agg * (w1[0,:]+w1[1,:])
    const float u0 = w1[0] + w1[4];
    const float u1 = w1[1] + w1[5];
    const float u2 = w1[2] + w1[6];
    const float u3 = w1[3] + w1[7];

    const float a_val = agg[base + m] + bc;
    const float uk0 = half ? u2    : u0;
    const float uk1 = half ? u3    : u1;
    const float bk0 = half ? b1[2] : b1[0];
    const float bk1 = half ? b1[3] : b1[1];

    v2f A;
    A.x = fmaxf(fmaf(a_val, uk0, bk0), 0.0f);   // h at K = 2*half
    A.y = fmaxf(fmaf(a_val, uk1, bk1), 0.0f);   // h at K = 2*half + 1
    // B (4x16 fp32): row K striped across lanes (N = lane%16); only N=0 nonzero.
    v2f B;
    B.x = (lane == 0) ? w2[0] : ((lane == 16) ? w2[2] : 0.0f);
    B.y = (lane == 0) ? w2[1] : ((lane == 16) ? w2[3] : 0.0f);
    v8f C = {};
    C = __builtin_amdgcn_wmma_f32_16x16x4_f32(
        /*neg_a=*/false, A, /*neg_b=*/false, B,
        /*c_mod=*/(short)0, C, /*reuse_a=*/false, /*reuse_b=*/false);

    const float bias2 = b2[0];
    if (m == 0) {                       // lanes 0 and 16 hold column N=0
        const int ob = base + half * 8; // lane 0 -> M 0..7, lane 16 -> M 8..15
        float4 o0 = make_float4(C[0] + bias2, C[1] + bias2, C[2] + bias2, C[3] + bias2);
        float4 o1 = make_float4(C[4] + bias2, C[5] + bias2, C[6] + bias2, C[7] + bias2);
        *(float4*)(out + ob)     = o0;
        *(float4*)(out + ob + 4) = o1;
    }
}

// ---------------------------------------------------------------------------
extern "C" void kernel_launch(void* const* d_in, const int* in_sizes, int n_in,
                              void* d_out, int out_size, void* d_ws, size_t ws_size,
                              hipStream_t stream) {
    const float*     x      = (const float*)d_in[0];
    const long long* eidx   = (const long long*)d_in[1];  // int64, shape (2, E) flat
    const float*     w_conv = (const float*)d_in[2];
    const float*     b_conv = (const float*)d_in[3];
    const float*     w1     = (const float*)d_in[4];
    const float*     b1     = (const float*)d_in[5];
    const float*     w2     = (const float*)d_in[6];
    const float*     b2     = (const float*)d_in[7];

    const long long* src = eidx;                 // row 0
    const long long* dst = eidx + GCN_N_EDGES;   // row 1

    float* agg = (float*)d_ws;                   // 400K floats = 1.6 MB
    float* out = (float*)d_out;

    gcn_zero_agg<<<(GCN_N_OUT + 255) / 256, 256, 0, stream>>>(agg);

    // 4096 blocks x 256 threads = 32768 wave32s; 16 edge-pairs per thread.
    gcn_edge_scatter<<<4096, 256, 0, stream>>>(src, dst, x, w_conv, agg);

    // exactly 25000 waves (400000/16 groups), 8 waves per block.
    gcn_mlp_wmma<<<3125, 256, 0, stream>>>(agg, b_conv, w1, b1, w2, b2, out);
}